// Model_39676907886828
// MI455X (gfx1250) — compile-verified
//
#include <hip/hip_runtime.h>

// CDNA5 / gfx1250: fp32 WMMA chain for a 2x2 matmul DAG.
// D = A(16x4) x B(4x16) + C(16x16), wave32, A/B = v2f, C/D = v8f.
// Cross-lane broadcasts use v_readlane (SGPR path) instead of ds_bpermute.

typedef __attribute__((ext_vector_type(2))) float v2f;
typedef __attribute__((ext_vector_type(8))) float v8f;

__device__ __forceinline__ v8f wmma_f32(v2f a, v2f b, v8f c) {
    // (neg_a, A, neg_b, B, c_mod, C, reuse_a, reuse_b)
    return __builtin_amdgcn_wmma_f32_16x16x4_f32(
        false, a, false, b, (short)0, c, false, false);
}

// Broadcast a float from a compile-time-constant lane via v_readlane_b32
// (uniform SGPR result; avoids the LDS crossbar / DScnt path of __shfl).
__device__ __forceinline__ float readlane_f(float v, int srcLane) {
    int r = __builtin_amdgcn_readlane(__builtin_bit_cast(int, v), srcLane);
    return __builtin_bit_cast(float, r);
}

// A-layout (16x4, 32-bit): VGPR k holds K=k (lanes 0-15), K=k+2 (lanes 16-31),
// lane indexes M. 2x2 tile -> lane0={m00,m01}, lane1={m10,m11}, rest 0.
__device__ __forceinline__ v2f makeA(float m00, float m01, float m10, float m11,
                                     int lane) {
    v2f a;
    a.x = (lane == 0) ? m00 : ((lane == 1) ? m10 : 0.0f);
    a.y = (lane == 0) ? m01 : ((lane == 1) ? m11 : 0.0f);
    return a;
}

// B-layout (4x16, 32-bit): VGPR k holds K=k (lanes 0-15), lane indexes N.
// 2x2 tile -> lane0={m00,m10}, lane1={m01,m11}, rest 0.
__device__ __forceinline__ v2f makeB(float m00, float m01, float m10, float m11,
                                     int lane) {
    v2f b;
    b.x = (lane == 0) ? m00 : ((lane == 1) ? m01 : 0.0f);
    b.y = (lane == 0) ? m10 : ((lane == 1) ? m11 : 0.0f);
    return b;
}

// Pull the 2x2 submatrix out of a D accumulator (VGPR0/1, lanes 0/1).
__device__ __forceinline__ void extractD(const v8f c, float& m00, float& m01,
                                         float& m10, float& m11) {
    m00 = readlane_f(c[0], 0);
    m01 = readlane_f(c[0], 1);
    m10 = readlane_f(c[1], 0);
    m11 = readlane_f(c[1], 1);
}

__global__ void __launch_bounds__(32)
chain2x2_wmma_kernel(const float* __restrict__ x3, const float* __restrict__ x4,
                     const float* __restrict__ x5, const float* __restrict__ x6,
                     const float* __restrict__ x7, float* __restrict__ out) {
    const int lane = (int)threadIdx.x;

    const float4 m3 = *reinterpret_cast<const float4*>(x3);
    const float4 m4 = *reinterpret_cast<const float4*>(x4);
    const float4 m5 = *reinterpret_cast<const float4*>(x5);
    const float4 m6 = *reinterpret_cast<const float4*>(x6);
    const float4 m7 = *reinterpret_cast<const float4*>(x7);

    const v8f zero = {};

    // v3 = x3 @ x7
    v2f A3 = makeA(m3.x, m3.y, m3.z, m3.w, lane);
    v2f B7 = makeB(m7.x, m7.y, m7.z, m7.w, lane);
    v8f v3 = wmma_f32(A3, B7, zero);

    // v6 = x6 @ x5
    v2f A6 = makeA(m6.x, m6.y, m6.z, m6.w, lane);
    v2f B5 = makeB(m5.x, m5.y, m5.z, m5.w, lane);
    v8f v6 = wmma_f32(A6, B5, zero);

    // D-layout of a 2x2 result doubles as its B-layout: reuse directly.
    v2f Bv6 = {v6[0], v6[1]};

    // v4 = x4 @ v6
    v2f A4 = makeA(m4.x, m4.y, m4.z, m4.w, lane);
    v8f v4 = wmma_f32(A4, Bv6, zero);

    // acc = v3 @ v6  (v3 as A operand needs the 2x2 transpose via readlane)
    float t00, t01, t10, t11;
    extractD(v3, t00, t01, t10, t11);
    v2f Av3 = makeA(t00, t01, t10, t11, lane);
    v8f acc = wmma_f32(Av3, Bv6, zero);

    // out = v4 @ x6 + acc   (final add folded into the WMMA accumulator)
    extractD(v4, t00, t01, t10, t11);
    v2f Av4 = makeA(t00, t01, t10, t11, lane);
    v2f B6  = makeB(m6.x, m6.y, m6.z, m6.w, lane);
    v8f res = wmma_f32(Av4, B6, acc);

    // D layout: res[0] lane n = out[0][n], res[1] lane n = out[1][n]
    if (lane < 2) {
        out[lane]     = res[0];
        out[2 + lane] = res[1];
    }
}

extern "C" void kernel_launch(void* const* d_in, const int* in_sizes, int n_in,
                              void* d_out, int out_size, void* d_ws, size_t ws_size,
                              hipStream_t stream) {
    // setup_inputs order: x1 x2 x3 x4 x5 x6 x7 ; x1, x2 unused by reference.
    const float* x3 = (const float*)d_in[2];
    const float* x4 = (const float*)d_in[3];
    const float* x5 = (const float*)d_in[4];
    const float* x6 = (const float*)d_in[5];
    const float* x7 = (const float*)d_in[6];
    float* out = (float*)d_out;

    chain2x2_wmma_kernel<<<1, 32, 0, stream>>>(x3, x4, x5, x6, x7, out);
}